// ZeGe_84250078478730
// MI455X (gfx1250) — compile-verified
//
#include <hip/hip_runtime.h>
#include <hip/hip_bf16.h>

typedef __attribute__((ext_vector_type(2))) float v2f;
typedef __attribute__((ext_vector_type(8))) float v8f;

#define DIM 64

// prop = 0.5 * item_feature   (residual half folded in; SpMM adds the other half)
__global__ void JQrx_prop_init(const float* __restrict__ feat,
                               float* __restrict__ prop, int n) {
    int i = blockIdx.x * blockDim.x + threadIdx.x;
    if (i < n) prop[i] = 0.5f * feat[i];
}

// COO scatter: prop[row] += 0.5 * val * feat[col]; 16 threads/edge, float4 each.
__global__ void JQrx_spmm_scatter(const float* __restrict__ feat,
                                  const float* __restrict__ vals,
                                  const int* __restrict__ rows,
                                  const int* __restrict__ cols,
                                  float* __restrict__ prop, int nE) {
    int t = blockIdx.x * blockDim.x + threadIdx.x;
    int e = t >> 4;
    if (e >= nE) return;
    int part = (t & 15) * 4;
    int c = cols[e];
    int r = rows[e];
    float v = 0.5f * vals[e];
    const float4 f = *(const float4*)(feat + (size_t)c * DIM + part);
    float* dst = prop + (size_t)r * DIM + part;
    atomicAdd(dst + 0, f.x * v);
    atomicAdd(dst + 1, f.y * v);
    atomicAdd(dst + 2, f.z * v);
    atomicAdd(dst + 3, f.w * v);
}

// aF = prop[idx]; wF = prop[i_pos] - prop[i_neg]; also zero the loss accumulator.
__global__ void JQrx_gather(const float* __restrict__ prop,
                            const int* __restrict__ idx,
                            const int* __restrict__ ipos,
                            const int* __restrict__ ineg,
                            float* __restrict__ aF, float* __restrict__ wF,
                            float* __restrict__ out, int B) {
    int t = blockIdx.x * blockDim.x + threadIdx.x;
    if (t == 0) out[0] = 0.0f;
    if (t >= B * DIM) return;
    int b = t / DIM, d = t % DIM;
    aF[t] = prop[(size_t)idx[b] * DIM + d];
    wF[t] = prop[(size_t)ipos[b] * DIM + d] - prop[(size_t)ineg[b] * DIM + d];
}

// Fused fp32-WMMA GEMM (diff = aF @ wF^T, 16x16 tiles, K=64 as 16 x k4 steps)
// + stable fast softplus(-x) + wave reduction + one atomic per wave.
// One wave (32 lanes) per block; each wave: 1 row-tile x 16 col-tiles.
// B-tile fragments are batch-loaded into registers first so the compiler can
// clause the 16 global_load_b64s and issue the 16 WMMAs back-to-back.
__global__ void __launch_bounds__(32)
JQrx_loss_wmma(const float* __restrict__ aF, const float* __restrict__ wF,
               float* __restrict__ out, float scale) {
    const int lane = threadIdx.x;      // 0..31
    const int mn   = lane & 15;        // M (for A) / N (for B)
    const int h    = lane >> 4;        // lane half selects K pair
    const int rowTile  = blockIdx.x;   // 16 rows each
    const int colGroup = blockIdx.y;   // 256 cols each

    // Preload all A fragments for this 16x64 row strip: 16 x v2f = 32 VGPRs.
    // ISA layout (32-bit A 16x4): lane L -> M=L%16, VGPR v -> K = 2*(L/16)+v.
    v2f afrag[16];
    {
        const float* arow = aF + (size_t)(rowTile * 16 + mn) * DIM;
        #pragma unroll
        for (int kk = 0; kk < 16; ++kk) {
            afrag[kk] = *(const v2f*)(arow + kk * 4 + h * 2);
        }
    }

    float partial = 0.0f;
    #pragma unroll 1
    for (int ct = 0; ct < 16; ++ct) {
        const int col = colGroup * 256 + ct * 16 + mn;
        const float* wrow = wF + (size_t)col * DIM;

        // Stage the whole B tile (16 x v2f = 32 VGPRs) -> one load clause.
        v2f bfrag[16];
        #pragma unroll
        for (int kk = 0; kk < 16; ++kk) {
            bfrag[kk] = *(const v2f*)(wrow + kk * 4 + h * 2);
        }

        // 16 back-to-back fp32 WMMAs accumulate the 16x16 tile over K=64.
        v8f acc = {};
        #pragma unroll
        for (int kk = 0; kk < 16; ++kk) {
            acc = __builtin_amdgcn_wmma_f32_16x16x4_f32(
                false, afrag[kk], false, bfrag[kk], (short)0, acc, false, false);
        }

        // -log(sigmoid(x)) = softplus(-x) = max(-x,0) + log(1 + exp(-|x|))
        // exp arg <= 0 and log arg in (1,2]: hardware v_exp/v_log are safe here.
        #pragma unroll
        for (int e2 = 0; e2 < 8; ++e2) {
            float x = acc[e2];
            partial += fmaxf(-x, 0.0f) + __logf(1.0f + __expf(-fabsf(x)));
        }
    }

    // wave32 reduction
    #pragma unroll
    for (int off = 16; off > 0; off >>= 1)
        partial += __shfl_down(partial, off, 32);
    if (lane == 0) atomicAdd(out, partial * scale);
}

extern "C" void kernel_launch(void* const* d_in, const int* in_sizes, int n_in,
                              void* d_out, int out_size, void* d_ws, size_t ws_size,
                              hipStream_t stream) {
    const float* feat  = (const float*)d_in[0];
    const float* evals = (const float*)d_in[1];
    const int*   erows = (const int*)d_in[2];
    const int*   ecols = (const int*)d_in[3];
    const int*   idx   = (const int*)d_in[4];
    const int*   ipos  = (const int*)d_in[5];
    const int*   ineg  = (const int*)d_in[6];
    float* out = (float*)d_out;

    const int nFeat = in_sizes[0];        // N * 64
    const int N     = nFeat / DIM;
    const int E     = in_sizes[1];
    const int B     = in_sizes[4];        // 4096

    float* prop = (float*)d_ws;                     // N x 64  (25.6 MB)
    float* aFp  = prop + (size_t)N * DIM;           // B x 64  (1 MB)
    float* wFp  = aFp  + (size_t)B * DIM;           // B x 64  (1 MB)

    JQrx_prop_init<<<(nFeat + 255) / 256, 256, 0, stream>>>(feat, prop, nFeat);

    long long scatterThreads = (long long)E * 16;
    JQrx_spmm_scatter<<<(int)((scatterThreads + 255) / 256), 256, 0, stream>>>(
        feat, evals, erows, ecols, prop, E);

    JQrx_gather<<<(B * DIM + 255) / 256, 256, 0, stream>>>(
        prop, idx, ipos, ineg, aFp, wFp, out, B);

    const float scale = 1.0f / ((float)B * (float)B);
    dim3 grid(B / 16, B / 256);   // 256 row-tiles x 16 col-groups for B=4096
    JQrx_loss_wmma<<<grid, 32, 0, stream>>>(aFp, wFp, out, scale);
}